// GraphSAGEEncoder_68865505624263
// MI455X (gfx1250) — compile-verified
//
#include <hip/hip_runtime.h>

typedef __attribute__((ext_vector_type(2))) float v2f;
typedef __attribute__((ext_vector_type(8))) float v8f;

#define NN 100000   // nodes
#define NE 1600000  // edges
#define FD 64       // DIN == HID == OUT

// ---------------------------------------------------------------- zero fill
__global__ void zero_kernel(float* __restrict__ p, long long n) {
    long long i = (long long)blockIdx.x * blockDim.x + threadIdx.x;
    long long stride = (long long)gridDim.x * blockDim.x;
    for (; i < n; i += stride) p[i] = 0.0f;
}

// ---------------------------------------------------------------- in-degree
__global__ void count_kernel(const long long* __restrict__ dst,
                             float* __restrict__ cnt) {
    int e = blockIdx.x * blockDim.x + threadIdx.x;
    if (e < NE) atomicAdd(&cnt[(int)dst[e]], 1.0f);
}

// ------------------------------------------------- 1 / max(cnt, 1) per node
__global__ void inv_kernel(const float* __restrict__ cnt,
                           float* __restrict__ inv) {
    int i = blockIdx.x * blockDim.x + threadIdx.x;
    if (i < NN) inv[i] = 1.0f / fmaxf(cnt[i], 1.0f);
}

// ------------------------------------------- scatter: agg[dst] += feat[src]
// one wave32 per edge, each lane owns 2 consecutive floats (float2 gather,
// two global_atomic_add_f32). x/h and agg live in L2 (192 MB) so these are
// L2-resident atomics, not HBM round-trips.
__global__ void scatter_kernel(const float* __restrict__ feat,
                               const long long* __restrict__ src,
                               const long long* __restrict__ dst,
                               float* __restrict__ agg) {
    long long tid = (long long)blockIdx.x * blockDim.x + threadIdx.x;
    long long e = tid >> 5;
    int lane = threadIdx.x & 31;
    if (e >= NE) return;
    int s = (int)src[e];
    int d = (int)dst[e];
    const float2 v = ((const float2*)(feat + (long long)s * FD))[lane];
    float* out = agg + (long long)d * FD + lane * 2;
    atomicAdd(out,     v.x);
    atomicAdd(out + 1, v.y);
}

// -------------------------------------------------------- fused SAGE layer
// out[m,n] = act( sum_k (agg[m,k]*inv[m]) * Wl[n,k]
//               + sum_k  xin[m,k]         * Wr[n,k]  + b[n] )
// One wave per 16x16 output tile; 4 waves/block cover the 64 output columns
// of 16 rows. K-loop in steps of 4 using V_WMMA_F32_16X16X4_F32 (f32-exact).
// N % 16 == 0 so EXEC is all-ones (WMMA requirement).
__global__ __launch_bounds__(128)
void sage_gemm_kernel(const float* __restrict__ agg,
                      const float* __restrict__ inv,
                      const float* __restrict__ xin,
                      const float* __restrict__ Wl,   // [64,64] (out,in)
                      const float* __restrict__ Wr,   // [64,64] (out,in)
                      const float* __restrict__ bias, // [64]
                      float* __restrict__ out,        // [N,64]
                      int do_relu) {
    const int wave = threadIdx.x >> 5;   // 0..3 -> column tile
    const int lane = threadIdx.x & 31;
    const int half = lane >> 4;          // 0: lanes 0-15, 1: lanes 16-31
    const int l16  = lane & 15;
    const int row0 = blockIdx.x * 16;
    const int n0   = wave * 16;
    const int m = row0 + l16;            // A-row this lane feeds
    const int n = n0 + l16;              // B-column this lane feeds

    const float scale = inv[m];
    const float* aggm = agg + (long long)m * FD;
    const float* xm   = xin + (long long)m * FD;
    const float* wln  = Wl  + (long long)n * FD;
    const float* wrn  = Wr  + (long long)n * FD;

    v8f acc = {};
#pragma unroll
    for (int k0 = 0; k0 < FD; k0 += 4) {
        const int ka = k0 + 2 * half;    // ISA A/B layout: lanes16-31 hold K+2,K+3
        v2f a, b;
        // mean-aggregated term: scale folded into A load
        a.x = aggm[ka]     * scale;
        a.y = aggm[ka + 1] * scale;
        b.x = wln[ka];                   // B[k][n] = Wl[n][k]
        b.y = wln[ka + 1];
        acc = __builtin_amdgcn_wmma_f32_16x16x4_f32(
            false, a, false, b, (short)0, acc, false, false);
        // self (root) term
        a.x = xm[ka];
        a.y = xm[ka + 1];
        b.x = wrn[ka];
        b.y = wrn[ka + 1];
        acc = __builtin_amdgcn_wmma_f32_16x16x4_f32(
            false, a, false, b, (short)0, acc, false, false);
    }

    const float bn = bias[n];
#pragma unroll
    for (int v = 0; v < 8; ++v) {
        // D layout: vgpr v, lanes 0-15 -> M=v, lanes 16-31 -> M=v+8; N=l16
        const int mrow = row0 + v + 8 * half;
        float val = acc[v] + bn;
        if (do_relu) val = fmaxf(val, 0.0f);
        out[(long long)mrow * FD + n] = val;
    }
}

// ---------------------------------------------------------------- launcher
extern "C" void kernel_launch(void* const* d_in, const int* in_sizes, int n_in,
                              void* d_out, int out_size, void* d_ws, size_t ws_size,
                              hipStream_t stream) {
    const float*     x   = (const float*)d_in[0];
    const long long* ei  = (const long long*)d_in[1];   // [2, E] int64
    const float*     W1l = (const float*)d_in[2];
    const float*     W1r = (const float*)d_in[3];
    const float*     b1  = (const float*)d_in[4];
    const float*     W2l = (const float*)d_in[5];
    const float*     W2r = (const float*)d_in[6];
    const float*     b2  = (const float*)d_in[7];
    float*           out = (float*)d_out;

    const long long* src = ei;
    const long long* dst = ei + NE;

    // workspace layout (floats): agg[N*64] | cnt[N] | inv[N] | h[N*64]
    float* agg = (float*)d_ws;
    float* cnt = agg + (long long)NN * FD;
    float* inv = cnt + NN;
    float* h   = inv + NN;

    const int ZB = 2048;                       // zero-fill blocks (grid-stride)
    const long long aggN = (long long)NN * FD;

    // 1. zero agg + cnt (contiguous)
    zero_kernel<<<ZB, 256, 0, stream>>>(agg, aggN + NN);
    // 2. in-degree
    count_kernel<<<(NE + 255) / 256, 256, 0, stream>>>(dst, cnt);
    // 3. scatter layer-1: agg[dst] += x[src]   (E waves)
    {
        long long threads = (long long)NE * 32;
        scatter_kernel<<<(unsigned)((threads + 255) / 256), 256, 0, stream>>>(
            x, src, dst, agg);
    }
    // 4. inverse degree
    inv_kernel<<<(NN + 255) / 256, 256, 0, stream>>>(cnt, inv);
    // 5. layer 1: h = relu(mean@W1l^T + x@W1r^T + b1)
    sage_gemm_kernel<<<NN / 16, 128, 0, stream>>>(agg, inv, x, W1l, W1r, b1, h, 1);
    // 6. re-zero agg for layer 2
    zero_kernel<<<ZB, 256, 0, stream>>>(agg, aggN);
    // 7. scatter layer-2: agg[dst] += h[src]
    {
        long long threads = (long long)NE * 32;
        scatter_kernel<<<(unsigned)((threads + 255) / 256), 256, 0, stream>>>(
            h, src, dst, agg);
    }
    // 8. layer 2: out = mean@W2l^T + h@W2r^T + b2
    sage_gemm_kernel<<<NN / 16, 128, 0, stream>>>(agg, inv, h, W2l, W2r, b2, out, 0);
}